// PointsetGrouper_46832323396214
// MI455X (gfx1250) — compile-verified
//
#include <hip/hip_runtime.h>
#include <hip/hip_bf16.h>

#define B_   8
#define N_   4096
#define D_   128
#define G_   1024       // N/REDUCE
#define K_   32
#define R2_  0.0225f    // 0.15^2
#define BIG_ 1e10f
#define QB_  16         // queries per block (one WMMA tile of rows)
#define CH_  128        // support points per chunk (8 WMMA col-tiles)

typedef float v2f __attribute__((ext_vector_type(2)));
typedef float v8f __attribute__((ext_vector_type(8)));

#if __has_builtin(__builtin_amdgcn_global_load_async_to_lds_b128)
#define HAVE_ASYNC_LDS 1
// Builtin expects: (v4i addrspace(1)* src, v4i addrspace(3)* dst, imm offset, imm cpol)
typedef int async_b128_t __attribute__((vector_size(16)));
typedef __attribute__((address_space(1))) async_b128_t* async_gptr_t;
typedef __attribute__((address_space(3))) async_b128_t* async_lptr_t;
#else
#define HAVE_ASYNC_LDS 0
#endif

__device__ __forceinline__ void wait_asynccnt0() {
#if __has_builtin(__builtin_amdgcn_s_wait_asynccnt)
  __builtin_amdgcn_s_wait_asynccnt(0);
#else
  asm volatile("s_wait_asynccnt 0x0" ::: "memory");
#endif
}

__device__ __forceinline__ unsigned ballot32(bool p) {
#if __has_builtin(__builtin_amdgcn_ballot_w32)
  return __builtin_amdgcn_ballot_w32(p);
#else
  return (unsigned)__ballot(p);
#endif
}

// ---------------------------------------------------------------------------
// Kernel 1: furthest point sampling. One 1024-thread block (32 waves) per batch.
// xyz staged in LDS (48KB); per-thread min-dists in 4 VGPRs; argmax via
// wave shuffle reduce + cross-wave LDS reduce with first-index tie-break.
// ---------------------------------------------------------------------------
__global__ __launch_bounds__(1024) void fps_kernel(const float* __restrict__ xyz,
                                                   float* __restrict__ out_xyz,
                                                   float* __restrict__ out_pts) {
  __shared__ float sx[N_], sy[N_], sz[N_];
  __shared__ float wval[32];
  __shared__ int   widx[32];
  __shared__ int   lastIdx;

  const int b   = blockIdx.x;
  const int tid = threadIdx.x;
  const float* base = xyz + (size_t)b * N_ * 3;

  for (int i = tid; i < N_; i += 1024) {
    sx[i] = base[i * 3 + 0];
    sy[i] = base[i * 3 + 1];
    sz[i] = base[i * 3 + 2];
  }
  if (tid == 0) lastIdx = 0;
  __syncthreads();

  float dist[4];
#pragma unroll
  for (int t = 0; t < 4; ++t) dist[t] = BIG_;

  int last = 0;
  for (int it = 0; it < G_; ++it) {
    const float px = sx[last], py = sy[last], pz = sz[last];
    if (tid == 0) {
      out_xyz[((size_t)b * G_ + it) * 3 + 0] = px;
      out_xyz[((size_t)b * G_ + it) * 3 + 1] = py;
      out_xyz[((size_t)b * G_ + it) * 3 + 2] = pz;
      out_pts[((size_t)b * 131 + 128) * G_ + it] = px;   // use_xyz concat rows
      out_pts[((size_t)b * 131 + 129) * G_ + it] = py;
      out_pts[((size_t)b * 131 + 130) * G_ + it] = pz;
    }
    float bv = -1.0f; int bi = 0;
#pragma unroll
    for (int t = 0; t < 4; ++t) {
      const int i = t * 1024 + tid;
      const float dx = sx[i] - px, dy = sy[i] - py, dz = sz[i] - pz;
      const float d  = dx * dx + dy * dy + dz * dz;
      dist[t] = fminf(dist[t], d);
      if (dist[t] > bv) { bv = dist[t]; bi = i; }   // ascending i: strict '>' keeps first max
    }
    // wave32 reduction (max value, min index on tie == jnp.argmax semantics)
#pragma unroll
    for (int off = 16; off > 0; off >>= 1) {
      const float ov = __shfl_down(bv, off);
      const int   oi = __shfl_down(bi, off);
      if (ov > bv || (ov == bv && oi < bi)) { bv = ov; bi = oi; }
    }
    const int wid = tid >> 5;
    if ((tid & 31) == 0) { wval[wid] = bv; widx[wid] = bi; }
    __syncthreads();
    if (tid < 32) {
      bv = wval[tid]; bi = widx[tid];
#pragma unroll
      for (int off = 16; off > 0; off >>= 1) {
        const float ov = __shfl_down(bv, off);
        const int   oi = __shfl_down(bi, off);
        if (ov > bv || (ov == bv && oi < bi)) { bv = ov; bi = oi; }
      }
      if (tid == 0) lastIdx = bi;
    }
    __syncthreads();
    last = lastIdx;
  }
}

// ---------------------------------------------------------------------------
// Kernel 2: ball query (WMMA distance tiles) + ordered first-k selection +
// gather + mean-center + channel max. 256 threads (8 waves) per 16 queries.
//
// WMMA trick: D[m][n] = sum_k A[m][k]*B[k][n] with
//   A[m] = [-2qx, -2qy, -2qz, 1],  B[:,n] = [sx, sy, sz, s2]  =>  s2 - 2 q.s
// and the in-ball test becomes  D[m][n] <= r^2 - q2[m].
// f32 16x16x4 operand layout (ISA 7.12.2):
//   A (16x4): lanes 0-15 -> M, v0={K0|K2}, v1={K1|K3} (upper half lanes K2/K3)
//   B (4x16): lanes%16 -> N, v0={K0|K2}, v1={K1|K3}
//   D (16x16): VGPR r, lane L -> M = r + 8*(L>=16), N = L%16
// ---------------------------------------------------------------------------
__global__ __launch_bounds__(256) void group_kernel(const float* __restrict__ xyz,
                                                    const float* __restrict__ points,
                                                    const float* __restrict__ alpha,
                                                    const float* __restrict__ beta,
                                                    const float* __restrict__ new_xyz,
                                                    float* __restrict__ out_pts) {
  __shared__ float qx[QB_], qy[QB_], qz[QB_], thr[QB_];
  __shared__ __align__(16) float rawxyz[CH_ * 3];   // async-staged interleaved chunk
  __shared__ float sxc[CH_], syc[CH_], szc[CH_], s2c[CH_];
  __shared__ float dmat[QB_][CH_ + 4];
  __shared__ int   nbr[QB_][K_];
  __shared__ int   cnt[QB_];

  const int tid  = threadIdx.x;
  const int lane = tid & 31;
  const int w    = tid >> 5;                    // 8 waves
  const int b    = blockIdx.x / (G_ / QB_);
  const int qb   = (blockIdx.x % (G_ / QB_)) * QB_;

  if (tid < QB_) {
    const float x = new_xyz[((size_t)b * G_ + qb + tid) * 3 + 0];
    const float y = new_xyz[((size_t)b * G_ + qb + tid) * 3 + 1];
    const float z = new_xyz[((size_t)b * G_ + qb + tid) * 3 + 2];
    qx[tid] = x; qy[tid] = y; qz[tid] = z;
    thr[tid] = R2_ - (x * x + y * y + z * z);
    cnt[tid] = 0;
  }
  __syncthreads();

  // A operand is constant across all chunks; branchless via pointer select.
  v2f av;
  {
    const int m = lane & 15;
    const float* a0 = (lane < 16) ? qx : qz;
    const float  s0 = -2.0f * a0[m];
    const float  s1 = (lane < 16) ? -2.0f * qy[m] : 1.0f;
    av.x = s0; av.y = s1;
  }

  const float* sbase = xyz + (size_t)b * N_ * 3;
  for (int chunk = 0; chunk < N_; chunk += CH_) {
#if HAVE_ASYNC_LDS
    // CDNA5 async copy: global -> LDS, 16B per lane, ASYNCcnt-tracked.
    if (tid < (CH_ * 3) / 4) {
      float* g = const_cast<float*>(sbase + (size_t)chunk * 3 + tid * 4);
      __builtin_amdgcn_global_load_async_to_lds_b128(
          (async_gptr_t)g,
          (async_lptr_t)&rawxyz[tid * 4],
          0, 0);
    }
    if (chunk + CH_ < N_ && tid < CH_)   // warm next chunk into cache
      __builtin_prefetch(sbase + (size_t)(chunk + CH_) * 3 + tid * 3, 0, 1);
    wait_asynccnt0();
    __syncthreads();
    if (tid < CH_) {
      const float x = rawxyz[tid * 3 + 0];
      const float y = rawxyz[tid * 3 + 1];
      const float z = rawxyz[tid * 3 + 2];
      sxc[tid] = x; syc[tid] = y; szc[tid] = z;
      s2c[tid] = x * x + y * y + z * z;
    }
#else
    if (tid < CH_) {
      const float x = sbase[(chunk + tid) * 3 + 0];
      const float y = sbase[(chunk + tid) * 3 + 1];
      const float z = sbase[(chunk + tid) * 3 + 2];
      sxc[tid] = x; syc[tid] = y; szc[tid] = z;
      s2c[tid] = x * x + y * y + z * z;
    }
    if (chunk + CH_ < N_ && tid < CH_)
      __builtin_prefetch(sbase + (size_t)(chunk + CH_) * 3 + tid * 3, 0, 1);
#endif
    __syncthreads();

    // each wave computes one 16x16 distance tile via WMMA (branchless B build)
    {
      const int cn = w * 16 + (lane & 15);
      const float* p0 = (lane < 16) ? sxc : szc;
      const float* p1 = (lane < 16) ? syc : s2c;
      v2f bv; bv.x = p0[cn]; bv.y = p1[cn];
      v8f acc = {0.f, 0.f, 0.f, 0.f, 0.f, 0.f, 0.f, 0.f};
      acc = __builtin_amdgcn_wmma_f32_16x16x4_f32(false, av, false, bv,
                                                  (short)0, acc, false, false);
      const int mrow = (lane >> 4) * 8;
#pragma unroll
      for (int r = 0; r < 8; ++r) dmat[mrow + r][cn] = acc[r];
    }
    __syncthreads();

    // ordered first-k selection; wave w owns query rows 2w and 2w+1
#pragma unroll
    for (int rr0 = 0; rr0 < 2; ++rr0) {
      const int rr = 2 * w + rr0;
      int c0 = cnt[rr];
      if (c0 < K_) {
        const float th = thr[rr];
        for (int itc = 0; itc < CH_ / 32 && c0 < K_; ++itc) {
          const int col  = itc * 32 + lane;
          const bool pred = dmat[rr][col] <= th;
          const unsigned mb = ballot32(pred);
          const int rank = __popc(mb & ((1u << lane) - 1u));
          const int slot = c0 + rank;
          if (pred && slot < K_) nbr[rr][slot] = chunk + col;
          c0 += __popc(mb);
          if (c0 > K_) c0 = K_;
        }
        if (lane == 0) cnt[rr] = c0;
      }
    }
    __syncthreads();
    bool alldone = true;
#pragma unroll
    for (int m2 = 0; m2 < QB_; ++m2) alldone = alldone && (cnt[m2] >= K_);
    if (alldone) break;              // uniform across block
  }

  __syncthreads();
  if (tid < QB_) {
    if (cnt[tid] == 0) { nbr[tid][0] = 0; cnt[tid] = 1; }   // defensive (query in own ball)
  }
  __syncthreads();
  for (int idx = tid; idx < QB_ * K_; idx += 256) {        // pad empties with first hit
    const int m = idx / K_, j = idx % K_;
    if (j >= cnt[m]) nbr[m][j] = nbr[m][0];
  }
  __syncthreads();

  // aggregation: 2 queries x 128 channels at a time; 8 rounds covers 16 queries
  const int   c  = tid & 127;
  const float al = alpha[c], be = beta[c];
  const int   qh = tid >> 7;                // 0 or 1
  const float* prow_base = points + (size_t)b * N_ * D_ + c;
  for (int qr = 0; qr < QB_; qr += 2) {
    const int q = qr + qh;
    float v[K_];
    float sum = 0.f;
#pragma unroll
    for (int j = 0; j < K_; ++j) {
      const int id = nbr[q][j];
      v[j] = prow_base[(size_t)id * D_];    // coalesced across channels; L2-resident
      sum += v[j];
    }
    const float mean = sum * (1.0f / K_);
    float mx = -3.402823466e38f;
#pragma unroll
    for (int j = 0; j < K_; ++j) {
      const float t = al * (v[j] - mean) + be;
      mx = fmaxf(mx, t);
    }
    out_pts[((size_t)b * 131 + c) * G_ + (qb + q)] = mx;
  }
}

extern "C" void kernel_launch(void* const* d_in, const int* in_sizes, int n_in,
                              void* d_out, int out_size, void* d_ws, size_t ws_size,
                              hipStream_t stream) {
  (void)in_sizes; (void)n_in; (void)out_size; (void)d_ws; (void)ws_size;
  const float* xyz    = (const float*)d_in[0];   // [8,4096,3]
  const float* points = (const float*)d_in[1];   // [8,4096,128]
  const float* alpha  = (const float*)d_in[2];   // [128]
  const float* beta   = (const float*)d_in[3];   // [128]

  float* out_xyz = (float*)d_out;                        // [8,1024,3]
  float* out_pts = out_xyz + (size_t)B_ * G_ * 3;        // [8,131,1024]

  fps_kernel<<<B_, 1024, 0, stream>>>(xyz, out_xyz, out_pts);
  group_kernel<<<B_ * (G_ / QB_), 256, 0, stream>>>(xyz, points, alpha, beta,
                                                    out_xyz, out_pts);
}